// SeSoap_50989851738335
// MI455X (gfx1250) — compile-verified
//
#include <hip/hip_runtime.h>
#include <hip/hip_bf16.h>
#include <math.h>

typedef __attribute__((ext_vector_type(2))) float v2f;
typedef __attribute__((ext_vector_type(8))) float v8f;

#define CHUNK  256   // atoms per block-iteration
#define STRIDE 57    // LDS row stride in floats (odd -> conflict-free banks)
#define NENT   294   // 6 radial * 49 harmonic entries
#define NTILES 19    // ceil(294/16)
#define NSPEC  3

// factorials 0!..11! for nnl
__device__ __constant__ float c_fact[12] = {
    1.f, 1.f, 2.f, 6.f, 24.f, 120.f, 720.f, 5040.f,
    40320.f, 362880.f, 3628800.f, 39916800.f};

__global__ __launch_bounds__(256) void soap_zero(float* cbuf) {
  int i = blockIdx.x * blockDim.x + threadIdx.x;
  if (i < NSPEC * NENT) cbuf[i] = 0.0f;
}

__global__ __launch_bounds__(256) void soap_accum(const float* __restrict__ coo,
                                                  const int* __restrict__ numbers,
                                                  float* __restrict__ cbuf,
                                                  int natoms, int nchunks) {
  __shared__ float sY[CHUNK * STRIDE];  // [atom][0..48]=Y(ij), [49..54]=f(n)
  __shared__ int   sSpec[CHUNK];

  const int tid  = threadIdx.x;
  const int lane = tid & 31;
  const int wave = tid >> 5;
  const int row  = lane & 15;           // M row within tile / N col for B
  const int kg   = (lane >> 4) << 1;    // K group: 0 or 2

  // Per-slot (tile) constants: entry e = tile*16 + row -> (n, ij).
  // Offsets clamped in-bounds; invalid lanes are zeroed via mask float.
  int   offF[3], offY[3];
  float msk[3];
#pragma unroll
  for (int s = 0; s < 3; ++s) {
    int tile = wave + 8 * s;
    int e = tile * 16 + row;
    bool v = (tile < NTILES) && (e < NENT);
    int n = e / 49; if (n > 5) n = 5;
    int ij = e - n * 49; if (ij > 48) ij = 48;
    offF[s] = 49 + n;
    offY[s] = ij;
    msk[s]  = v ? 1.0f : 0.0f;
  }
  // wave-uniform flag for slot 2 (tiles 16..18 exist only for waves 0..2)
  const bool haveSlot2 = (__builtin_amdgcn_readfirstlane(wave) < 3);

  v8f acc0 = {0.f, 0.f, 0.f, 0.f, 0.f, 0.f, 0.f, 0.f};
  v8f acc1 = acc0;
  v8f acc2 = acc0;

  for (int chunk = blockIdx.x; chunk < nchunks; chunk += gridDim.x) {
    const int base = chunk * CHUNK;
    __syncthreads();  // previous phase-B reads done before overwriting LDS

    // ---------------- Phase A: per-atom basis into LDS ----------------
    {
      int atom = base + tid;
      float x = 0.f, y = 0.f, z = 0.f;
      int spec = 0;
      if (atom < natoms) {
        x = coo[3 * atom + 0];
        y = coo[3 * atom + 1];
        z = coo[3 * atom + 2];
        int num = numbers[atom];           // species sorted: 1, 6, 8
        spec = (num < 6) ? 0 : ((num == 6) ? 1 : 2);
      }
      float r2 = x * x + y * y + z * z;
      float d  = sqrtf(r2);
      float t  = fmaxf(1.0f - d * (1.0f / 6.0f), 0.0f);   // RC = 6
      float rad = t * t * expf(-0.5f * r2);
      float* my = &sY[tid * STRIDE];
      float fk = rad;
#pragma unroll
      for (int n = 0; n < 6; ++n) { my[49 + n] = fk; fk *= r2; }

      // solid harmonics recurrence, fully unrolled -> registers
      float Rre[7][7], Rim[7][7];
      Rre[0][0] = 0.28209479177387814f;  // 1/sqrt(4*pi)
      Rim[0][0] = 0.0f;
#pragma unroll
      for (int l = 1; l < 7; ++l) {
        float fl = -sqrtf((2.0f * l + 1.0f) / (2.0f * l));
        Rre[l][l] = fl * (x * Rre[l-1][l-1] - y * Rim[l-1][l-1]);
        Rim[l][l] = fl * (x * Rim[l-1][l-1] + y * Rre[l-1][l-1]);
        float g = sqrtf(2.0f * l + 1.0f);
        Rre[l][l-1] = g * z * Rre[l-1][l-1];
        Rim[l][l-1] = g * z * Rim[l-1][l-1];
#pragma unroll
        for (int m = 0; m + 1 < l; ++m) {
          float a = sqrtf((4.0f * l * l - 1.0f) / (float)(l * l - m * m));
          float b = sqrtf(((l - 1.0f) * (l - 1.0f) - (float)(m * m)) /
                          (4.0f * (l - 1.0f) * (l - 1.0f) - 1.0f));
          Rre[l][m] = a * (z * Rre[l-1][m] - b * r2 * Rre[l-2][m]);
          Rim[l][m] = a * (z * Rim[l-1][m] - b * r2 * Rim[l-2][m]);
        }
      }
      // pack: Re at [l, l-m], Im at [l-m, l]
#pragma unroll
      for (int l = 0; l < 7; ++l) {
#pragma unroll
        for (int m = 0; m <= l; ++m) {
          my[l * 7 + (l - m)] = Rre[l][m];
          if (m > 0) my[(l - m) * 7 + l] = Rim[l][m];
        }
      }
      sSpec[tid] = spec;
    }
    __syncthreads();

    // ---------------- Phase B: WMMA f32 16x16x4 accumulation ----------------
#pragma unroll 2
    for (int kb = 0; kb < CHUNK; kb += 4) {
      const int aA = kb + kg;                 // K = kg     -> v0
      const float* pA = &sY[aA * STRIDE];
      const float* pB = pA + STRIDE;          // K = kg + 1 -> v1
      // Issue every LDS load for this step up-front (single dscnt wait).
      const int sA = sSpec[aA];
      const int sB = sSpec[aA + 1];
      const float fA0 = pA[offF[0]], yA0 = pA[offY[0]];
      const float fB0 = pB[offF[0]], yB0 = pB[offY[0]];
      const float fA1 = pA[offF[1]], yA1 = pA[offY[1]];
      const float fB1 = pB[offF[1]], yB1 = pB[offY[1]];

      v2f bf;
      bf.x = (sA == row) ? 1.0f : 0.0f;       // one-hot species; cols >=3 zero
      bf.y = (sB == row) ? 1.0f : 0.0f;

      v2f af;
      af.x = fA0 * yA0 * msk[0];
      af.y = fB0 * yB0 * msk[0];
      acc0 = __builtin_amdgcn_wmma_f32_16x16x4_f32(false, af, false, bf,
                                                   (short)0, acc0, false, false);
      af.x = fA1 * yA1 * msk[1];
      af.y = fB1 * yB1 * msk[1];
      acc1 = __builtin_amdgcn_wmma_f32_16x16x4_f32(false, af, false, bf,
                                                   (short)0, acc1, false, false);
      if (haveSlot2) {                        // scalar branch, EXEC untouched
        const float fA2 = pA[offF[2]], yA2 = pA[offY[2]];
        const float fB2 = pB[offF[2]], yB2 = pB[offY[2]];
        af.x = fA2 * yA2 * msk[2];
        af.y = fB2 * yB2 * msk[2];
        acc2 = __builtin_amdgcn_wmma_f32_16x16x4_f32(false, af, false, bf,
                                                     (short)0, acc2, false, false);
      }
    }
  }

  // ---------------- scatter D fragments to global partials ----------------
  const int hi = (lane >> 4) << 3;  // lanes 16-31 hold M = r + 8
#pragma unroll
  for (int s = 0; s < 3; ++s) {
    int tile = wave + 8 * s;
    v8f a = (s == 0) ? acc0 : (s == 1) ? acc1 : acc2;
#pragma unroll
    for (int r = 0; r < 8; ++r) {
      int entry = tile * 16 + r + hi;
      if (row < NSPEC && tile < NTILES && entry < NENT)
        atomicAdd(&cbuf[row * NENT + entry], a[r]);
    }
  }
}

__global__ __launch_bounds__(256) void soap_finalize(const float* __restrict__ cbuf,
                                                     float* __restrict__ out) {
  __shared__ float c[NSPEC * NENT];   // 882
  __shared__ float pbuf[2268];
  __shared__ float red[256];
  const int tid = threadIdx.x;

  for (int i = tid; i < NSPEC * NENT; i += 256) c[i] = cbuf[i];
  __syncthreads();

  float sumsq = 0.0f;
  for (int t = tid; t < 2268; t += 256) {
    int l  = t % 7;
    int q  = t / 7;
    int n1 = q % 6;  q /= 6;
    int n2 = q % 6;  q /= 6;
    int s1 = q % 3;
    int s2 = q / 3;
    const float* c1 = &c[(s1 * 6 + n1) * 49];
    const float* c2 = &c[(s2 * 6 + n2) * 49];
    // sum_j Yr[l,j]*c1[l,j]c2[l,j] + sum_i Yi[i,l]*c1[i,l]c2[i,l]
    float acc = c1[l * 7 + l] * c2[l * 7 + l];
    for (int j = 0; j < l; ++j) acc += 2.0f * c1[l * 7 + j] * c2[l * 7 + j];
    for (int i = 0; i < l; ++i) acc += 2.0f * c1[i * 7 + l] * c2[i * 7 + l];
    // nnl[n2,n1,l] = sqrt(a[n2,l]*a[n1,l]); a[n,l] = 1/((2l+1)*2^(2n+l)*n!*(n+l)!)
    float a1 = 1.0f / ((float)(2 * l + 1) * exp2f((float)(2 * n1 + l)) *
                       c_fact[n1] * c_fact[n1 + l]);
    float a2 = 1.0f / ((float)(2 * l + 1) * exp2f((float)(2 * n2 + l)) *
                       c_fact[n2] * c_fact[n2 + l]);
    float v = acc * sqrtf(a1 * a2);
    pbuf[t] = v;
    sumsq += v * v;
  }
  red[tid] = sumsq;
  __syncthreads();
  for (int s = 128; s > 0; s >>= 1) {
    if (tid < s) red[tid] += red[tid + s];
    __syncthreads();
  }
  float inv = 1.0f / (sqrtf(red[0]) + 1.1920929e-7f);

  // ab: row0 = [1,6,8]*3, row1 = [1,1,1,6,6,6,8,8,8]
  if (tid < 18) {
    int i = (tid < 9) ? (tid % 3) : ((tid - 9) / 3);
    out[tid] = (i == 0) ? 1.0f : (i == 1) ? 6.0f : 8.0f;
  }
  for (int t = tid; t < 2268; t += 256) out[18 + t] = pbuf[t] * inv;
}

extern "C" void kernel_launch(void* const* d_in, const int* in_sizes, int n_in,
                              void* d_out, int out_size, void* d_ws, size_t ws_size,
                              hipStream_t stream) {
  const float* coo     = (const float*)d_in[0];
  const int*   numbers = (const int*)d_in[1];
  float* cbuf = (float*)d_ws;       // 882 floats of partial c
  float* out  = (float*)d_out;      // 18 (ab) + 2268 (p)
  const int natoms  = in_sizes[1];
  int nchunks = (natoms + CHUNK - 1) / CHUNK;
  if (nchunks < 1) nchunks = 1;
  int grid = nchunks < 608 ? nchunks : 608;

  soap_zero<<<4, 256, 0, stream>>>(cbuf);
  soap_accum<<<grid, 256, 0, stream>>>(coo, numbers, cbuf, natoms, nchunks);
  soap_finalize<<<1, 256, 0, stream>>>(cbuf, out);
}